// SMPLRenderer_76699525972678
// MI455X (gfx1250) — compile-verified
//
#include <hip/hip_runtime.h>
#include <hip/hip_bf16.h>

#define NFACE   13776
#define NCHUNK  (NFACE/16)        // 861 exactly
#define RES     128
#define NPIX    (RES*RES)
#define NEAR_P  0.1f
#define FAR_P   25.0f
#define INV_FAR (1.0f/25.0f)
#define INV_NEAR (1.0f/0.1f)
#define CAMZ    2.7320508075688772f   /* -EYE.z = 1/tan(30deg)+1 */

typedef __attribute__((ext_vector_type(2))) float v2f;
typedef __attribute__((ext_vector_type(8))) float v8f;

// Workspace layout (floats):
//   [ 0,  2NF)  AB0 pairs (a0,b0)      -> B operand K0/K1 rows for w0
//   [2NF, 4NF)  C0  pairs (c0, 0)      -> B operand K2/K3 rows for w0
//   [4NF, 6NF)  AB1 pairs (a1,b1)
//   [6NF, 8NF)  C1  pairs (c1, 0)
//   [8NF,10NF)  ABZ pairs (az,bz)
//  [10NF,12NF)  CZ  pairs (cz, 0)
//  [12NF,13NF)  Z0   [13NF,14NF) Z1   [14NF,15NF) Z2
// Total 15*NFACE floats = 826,560 bytes.

__global__ __launch_bounds__(256) void face_setup_kernel(
    const float* __restrict__ verts, const int* __restrict__ fidx,
    float* __restrict__ ws)
{
    int f = blockIdx.x * blockDim.x + threadIdx.x;
    if (f >= NFACE) return;
    int i0 = fidx[3*f+0], i1 = fidx[3*f+1], i2 = fidx[3*f+2];
    float x0 = verts[3*i0+0], y0 = verts[3*i0+1], z0 = verts[3*i0+2] + CAMZ;
    float x1 = verts[3*i1+0], y1 = verts[3*i1+1], z1 = verts[3*i1+2] + CAMZ;
    float x2 = verts[3*i2+0], y2 = verts[3*i2+1], z2 = verts[3*i2+2] + CAMZ;

    float det    = (y1-y2)*(x0-x2) + (x2-x1)*(y0-y2);
    bool  det_ok = fabsf(det) > 1e-10f;
    float dets   = det_ok ? det : 1e-10f;
    float inv    = 1.0f / dets;
    // w0 = a0*px + b0*py + c0 ; w1 = a1*px + b1*py + c1 ; w2 = 1-w0-w1
    float a0 = (y1-y2)*inv, b0 = (x2-x1)*inv;
    float c0 = -(a0*x2 + b0*y2);
    float a1 = (y2-y0)*inv, b1 = (x0-x2)*inv;
    float c1 = -(a1*x2 + b1*y2);

    bool ok = det_ok && (z0 > NEAR_P) && (z1 > NEAR_P) && (z2 > NEAR_P);
    float az = 0.0f, bz = 0.0f, cz = 0.0f;
    if (ok) {
        float iz0 = 1.0f/z0, iz1 = 1.0f/z1, iz2 = 1.0f/z2;
        float d0 = iz0 - iz2, d1 = iz1 - iz2;
        // inv_z = w0*iz0 + w1*iz1 + (1-w0-w1)*iz2 = az*px + bz*py + cz
        az = a0*d0 + a1*d1;
        bz = b0*d0 + b1*d1;
        cz = c0*d0 + c1*d1 + iz2;
    } else {
        // Sabotage: w0 always hugely negative -> never inside -> never wins.
        a0 = 0.0f; b0 = 0.0f; c0 = -1e30f;
        a1 = 0.0f; b1 = 0.0f; c1 = 0.0f;
    }

    v2f* AB0 = (v2f*)(ws + 0*NFACE);
    v2f* C0p = (v2f*)(ws + 2*NFACE);
    v2f* AB1 = (v2f*)(ws + 4*NFACE);
    v2f* C1p = (v2f*)(ws + 6*NFACE);
    v2f* ABZ = (v2f*)(ws + 8*NFACE);
    v2f* CZp = (v2f*)(ws + 10*NFACE);
    AB0[f] = (v2f){a0, b0};
    C0p[f] = (v2f){c0, 0.0f};
    AB1[f] = (v2f){a1, b1};
    C1p[f] = (v2f){c1, 0.0f};
    ABZ[f] = (v2f){az, bz};
    CZp[f] = (v2f){cz, 0.0f};
    ws[12*NFACE + f] = z0;
    ws[13*NFACE + f] = z1;
    ws[14*NFACE + f] = z2;
}

// One wave rasterizes 16 consecutive (row-major) pixels against all faces,
// 16 faces at a time via three V_WMMA_F32_16X16X4_F32 ops:
//   A (16x4): pixel rows [px, py, 1, 0]       (lanes 0-15: K0/K1, lanes 16-31: K2/K3)
//   B (4x16): face coeff columns [a; b; c; 0] (lanes 0-15: K0/K1, lanes 16-31: K2/K3)
//   D (16x16): w0 / w1 / inv_z for every (pixel, face) pair.
__global__ __launch_bounds__(256) void raster_kernel(
    const float* __restrict__ ws, const float* __restrict__ tex,
    float* __restrict__ out)
{
    __shared__ float s_d[8 * 16];
    __shared__ int   s_f[8 * 16];

    const int lane = threadIdx.x & 31;
    const int wave = threadIdx.x >> 5;
    const int g    = blockIdx.x * 8 + wave;   // 0..1023
    const int pix0 = g * 16;
    const int row  = pix0 >> 7;
    const int col0 = pix0 & 127;
    const int l16  = lane & 15;
    const bool hi  = lane >= 16;

    const float py = -((2.0f*(float)row + 1.0f - (float)RES) * (1.0f/(float)RES));
    const float px = (2.0f*(float)(col0 + l16) + 1.0f - (float)RES) * (1.0f/(float)RES);

    // A operand: lanes 0-15 -> (px, py); lanes 16-31 -> (1, 0)
    v2f A;
    A.x = hi ? 1.0f : px;
    A.y = hi ? 0.0f : py;

    // Per-lane B base pointers (pair-indexed): low lanes read (a,b), high lanes (c,0)
    const v2f* pb0 = (const v2f*)(ws + (hi ?  2*NFACE : 0       )) + l16;
    const v2f* pb1 = (const v2f*)(ws + (hi ?  6*NFACE : 4*NFACE)) + l16;
    const v2f* pbz = (const v2f*)(ws + (hi ? 10*NFACE : 8*NFACE)) + l16;

    // Running argmin in inverse-z space: bigger clipped inv_z == smaller depth.
    float bestv[8];
    int   bf[8];
#pragma unroll
    for (int r = 0; r < 8; ++r) { bestv[r] = INV_FAR; bf[r] = 0; }

    const v8f czero = {0.f,0.f,0.f,0.f,0.f,0.f,0.f,0.f};

    for (int c = 0; c < NCHUNK; ++c) {
        v2f B0 = pb0[c * 16];
        v2f B1 = pb1[c * 16];
        v2f BZ = pbz[c * 16];

        v8f W0 = __builtin_amdgcn_wmma_f32_16x16x4_f32(
            false, A, false, B0, (short)0, czero, false, false);
        v8f W1 = __builtin_amdgcn_wmma_f32_16x16x4_f32(
            false, A, false, B1, (short)0, czero, false, false);
        v8f WZ = __builtin_amdgcn_wmma_f32_16x16x4_f32(
            false, A, false, BZ, (short)0, czero, false, false);

        const int face = c * 16 + l16;
#pragma unroll
        for (int r = 0; r < 8; ++r) {
            float w0 = W0[r], w1 = W1[r], iz = WZ[r];
            float w2 = 1.0f - w0 - w1;
            float m  = fminf(fminf(w0, w1), w2);
            float izc  = fminf(iz, INV_NEAR);
            float cand = (m >= 0.0f) ? izc : 0.0f;   // invalid -> never beats INV_FAR
            bool upd = cand > bestv[r];              // strict: first face wins ties
            bestv[r] = upd ? cand : bestv[r];
            bf[r]    = upd ? face : bf[r];
        }
    }

    // Reduce over the 16 face-columns (lanes within each half) per pixel row r.
#pragma unroll
    for (int r = 0; r < 8; ++r) {
        float d = bestv[r];
        int   f = bf[r];
#pragma unroll
        for (int m = 1; m < 16; m <<= 1) {
            float od = __shfl_xor(d, m, 32);
            int   of = __shfl_xor(f, m, 32);
            bool take = (od > d) || ((od == d) && (of < f));
            d = take ? od : d;
            f = take ? of : f;
        }
        if (l16 == r) {                      // one writer per (half, r)
            int p = r + (hi ? 8 : 0);        // local pixel 0..15
            s_d[wave * 16 + p] = d;
            s_f[wave * 16 + p] = f;
        }
    }
    __syncthreads();

    if (lane < 16) {
        const int p      = lane;
        const float iz   = s_d[wave * 16 + p];
        const int   f    = s_f[wave * 16 + p];
        const int col    = col0 + p;
        const int pixidx = row * RES + col;
        const bool hit   = iz > INV_FAR;

        float rr = 0.0f, gg = 0.0f, bb = 0.0f;
        if (hit) {
            const float mind = 1.0f / iz;     // iz already clipped to [1/FAR, 1/NEAR]
            const float pxp  = (2.0f*(float)col + 1.0f - (float)RES) * (1.0f/(float)RES);
            v2f ab0   = ((const v2f*)(ws + 0*NFACE))[f];
            float c0  = ws[2*NFACE + 2*f];
            v2f ab1   = ((const v2f*)(ws + 4*NFACE))[f];
            float c1  = ws[6*NFACE + 2*f];
            float z0  = ws[12*NFACE + f];
            float z1  = ws[13*NFACE + f];
            float z2  = ws[14*NFACE + f];
            float w0  = ab0.x*pxp + ab0.y*py + c0;
            float w1  = ab1.x*pxp + ab1.y*py + c1;
            float w2  = 1.0f - w0 - w1;
            int t0 = (int)floorf(w0 / z0 * mind * 3.0f);
            int t1 = (int)floorf(w1 / z1 * mind * 3.0f);
            int t2 = (int)floorf(w2 / z2 * mind * 3.0f);
            t0 = min(max(t0, 0), 2);
            t1 = min(max(t1, 0), 2);
            t2 = min(max(t2, 0), 2);
            int lin = ((f * 3 + t0) * 3 + t1) * 3 + t2;   // == f*27 + t0*9 + t1*3 + t2
            rr = tex[lin * 3 + 0];
            gg = tex[lin * 3 + 1];
            bb = tex[lin * 3 + 2];
        }
        out[0 * NPIX + pixidx] = rr;          // imgs, CHW
        out[1 * NPIX + pixidx] = gg;
        out[2 * NPIX + pixidx] = bb;
        out[3 * NPIX + pixidx] = hit ? 1.0f : 0.0f;   // silhouette
    }
}

extern "C" void kernel_launch(void* const* d_in, const int* in_sizes, int n_in,
                              void* d_out, int out_size, void* d_ws, size_t ws_size,
                              hipStream_t stream) {
    const float* verts = (const float*)d_in[0];   // (1, 6890, 3) f32
    const float* tex   = (const float*)d_in[1];   // (1, 13776, 3,3,3, 3) f32
    const int*   faces = (const int*)d_in[2];     // (13776, 3) i32
    float* ws  = (float*)d_ws;                    // needs 15*NFACE floats (~827 KB)
    float* out = (float*)d_out;                   // 4*128*128 floats (imgs ++ sil)

    // Pass 1: per-face edge/inv-z coefficients (13776 faces)
    face_setup_kernel<<<dim3((NFACE + 255) / 256), dim3(256), 0, stream>>>(verts, faces, ws);

    // Pass 2: 1024 waves (16 px each) x 861 face chunks via WMMA 16x16x4 f32
    raster_kernel<<<dim3(NPIX / 16 / 8), dim3(256), 0, stream>>>(ws, tex, out);
}